// DengueGNN_15418932593124
// MI455X (gfx1250) — compile-verified
//
#include <hip/hip_runtime.h>
#include <hip/hip_bf16.h>
#include <math.h>

// ---------------------------------------------------------------------------
// DengueGNN: 2-layer GAT + linear head for MI455X (gfx1250, wave32)
// GEMMs use V_WMMA_F32_16X16X4_F32 (fp32-exact tensor path); edge softmax /
// aggregation uses encoded-float atomicMax + GLOBAL_ATOMIC_ADD_F32.
// ---------------------------------------------------------------------------

#define GNN_N_NODES 100000
#define GNN_N_EDGES 1600000
#define GNN_F_IN    64
#define GNN_HID     32
#define GNN_HEADS   4
#define GNN_NEG     0.2f
#define GNN_EPS     1e-16f

typedef __attribute__((ext_vector_type(2))) float v2f;
typedef __attribute__((ext_vector_type(8))) float v8f;

// Monotonic order-preserving float <-> uint encoding (handles negatives/-inf)
__device__ __forceinline__ unsigned fenc(float f) {
    unsigned u = __float_as_uint(f);
    return (u & 0x80000000u) ? ~u : (u | 0x80000000u);
}
__device__ __forceinline__ float fdec(unsigned u) {
    return __uint_as_float((u & 0x80000000u) ? (u & 0x7FFFFFFFu) : ~u);
}
#define FENC_NEG_INF 0x007FFFFFu  // fenc(-inf)

__device__ __forceinline__ float leaky(float v) {
    return v > 0.0f ? v : GNN_NEG * v;
}

// ---------------------------------------------------------------------------
// fp32 GEMM, one wave per 16x16 tile of C = A[M,K] x B[K,N], K%4==0, N%16==0,
// M%16==0.  Uses V_WMMA_F32_16X16X4_F32 stepping K by 4.
// A frag (16x4 f32, 2 VGPR): lane L holds A[L%16][2*(L/16)+{0,1}]
// B frag (4x16 f32, 2 VGPR): lane L holds B[2*(L/16)+{0,1}][L%16]
// C/D (16x16 f32, 8 VGPR):   vgpr r, lane L -> C[r + 8*(L/16)][L%16]
// ---------------------------------------------------------------------------
__global__ void gnn_gemm_wmma_f32(const float* __restrict__ A,
                                  const float* __restrict__ B,
                                  float* __restrict__ C,
                                  int M, int K, int N) {
    int wave  = (int)((blockIdx.x * blockDim.x + threadIdx.x) >> 5);
    int ntn   = N >> 4;
    int total = (M >> 4) * ntn;
    if (wave >= total) return;               // wave-uniform: EXEC stays all-1s
    int mt = wave / ntn;
    int nt = wave - mt * ntn;
    int lane = threadIdx.x & 31;
    int l16  = lane & 15;
    int half = lane >> 4;

    const float* Arow = A + (size_t)(mt * 16 + l16) * K;
    const float* Bcol = B + (nt * 16 + l16);

    v8f acc = {};
    for (int k0 = 0; k0 < K; k0 += 4) {
        int ka = k0 + half * 2;
        v2f a, b;
        a.x = Arow[ka];
        a.y = Arow[ka + 1];
        b.x = Bcol[(size_t)ka * N];
        b.y = Bcol[(size_t)(ka + 1) * N];
        acc = __builtin_amdgcn_wmma_f32_16x16x4_f32(
            /*neg_a=*/false, a, /*neg_b=*/false, b,
            /*c_mod=*/(short)0, acc, /*reuse_a=*/false, /*reuse_b=*/false);
    }

    float* Crow = C + (size_t)(mt * 16 + half * 8) * N + nt * 16 + l16;
#pragma unroll
    for (int r = 0; r < 8; ++r) Crow[(size_t)r * N] = acc[r];
}

// alpha_src[n,h] = <h[n,h,:], a_src[h,:]>, alpha_dst likewise
__global__ void gnn_attn_coef(const float* __restrict__ h,
                              const float* __restrict__ a_src,
                              const float* __restrict__ a_dst,
                              float* __restrict__ as, float* __restrict__ ad,
                              int n_nodes, int heads, int ch) {
    int t = blockIdx.x * blockDim.x + threadIdx.x;
    if (t >= n_nodes * heads) return;
    int n  = t / heads;
    int hh = t - n * heads;
    const float* hv = h + (size_t)n * heads * ch + (size_t)hh * ch;
    const float* sv = a_src + (size_t)hh * ch;
    const float* dv = a_dst + (size_t)hh * ch;
    float accs = 0.f, accd = 0.f;
    for (int c = 0; c < ch; ++c) {
        float v = hv[c];
        accs += v * sv[c];
        accd += v * dv[c];
    }
    as[t] = accs;
    ad[t] = accd;
}

// Re-init softmax state every launch (graph-replay safe)
__global__ void gnn_init_state(unsigned* __restrict__ menc, float* __restrict__ den,
                               float* __restrict__ agg, int nh, int nagg) {
    int t = blockIdx.x * blockDim.x + threadIdx.x;
    if (t < nh) { menc[t] = FENC_NEG_INF; den[t] = 0.0f; }
    if (t < nagg) agg[t] = 0.0f;
}

// Pass 1: segment max over incoming edges (incl. self loops appended at end)
__global__ void gnn_edge_max(const int* __restrict__ esrc, const int* __restrict__ edst,
                             int n_edges, int n_nodes, int heads,
                             const float* __restrict__ as, const float* __restrict__ ad,
                             unsigned* __restrict__ menc) {
    int t = blockIdx.x * blockDim.x + threadIdx.x;
    int total = (n_edges + n_nodes) * heads;
    if (t >= total) return;
    int e  = t / heads;
    int hh = t - e * heads;
    int s, d;
    if (e < n_edges) { s = esrc[e]; d = edst[e]; } else { s = d = e - n_edges; }
    float v = leaky(as[s * heads + hh] + ad[d * heads + hh]);
    atomicMax(&menc[d * heads + hh], fenc(v));
}

// Pass 2: denom = segment_sum(exp(e - m))
__global__ void gnn_edge_denom(const int* __restrict__ esrc, const int* __restrict__ edst,
                               int n_edges, int n_nodes, int heads,
                               const float* __restrict__ as, const float* __restrict__ ad,
                               const unsigned* __restrict__ menc, float* __restrict__ den) {
    int t = blockIdx.x * blockDim.x + threadIdx.x;
    int total = (n_edges + n_nodes) * heads;
    if (t >= total) return;
    int e  = t / heads;
    int hh = t - e * heads;
    int s, d;
    if (e < n_edges) { s = esrc[e]; d = edst[e]; } else { s = d = e - n_edges; }
    int idx = d * heads + hh;
    float v  = leaky(as[s * heads + hh] + ad[idx]);
    float ex = __expf(v - fdec(menc[idx]));
    atomicAdd(&den[idx], ex);
}

// Pass 3 (layer 1, H=4, C=32): one wave per edge; lane -> float4 of h[src]
__global__ void gnn_edge_agg_h4(const int* __restrict__ esrc, const int* __restrict__ edst,
                                int n_edges, int n_nodes,
                                const float* __restrict__ h,
                                const float* __restrict__ as, const float* __restrict__ ad,
                                const unsigned* __restrict__ menc,
                                const float* __restrict__ den,
                                float* __restrict__ out) {
    int wid = (int)((blockIdx.x * blockDim.x + threadIdx.x) >> 5);
    int total = n_edges + n_nodes;
    if (wid >= total) return;
    int lane = threadIdx.x & 31;
    int s, d;
    if (wid < n_edges) { s = esrc[wid]; d = edst[wid]; } else { s = d = wid - n_edges; }
    int hh  = lane >> 3;                       // 8 lanes (32 ch) per head
    int idx = d * GNN_HEADS + hh;
    float e = leaky(as[s * GNN_HEADS + hh] + ad[idx]);
    float alpha = __expf(e - fdec(menc[idx])) / (den[idx] + GNN_EPS);

    const float4 v = ((const float4*)(h + (size_t)s * 128))[lane];
    float* o = out + (size_t)d * 128 + lane * 4;
    atomicAdd(o + 0, v.x * alpha);
    atomicAdd(o + 1, v.y * alpha);
    atomicAdd(o + 2, v.z * alpha);
    atomicAdd(o + 3, v.w * alpha);
}

// Pass 3 (layer 2, H=1, C=32): one wave per edge; lane = channel
__global__ void gnn_edge_agg_h1(const int* __restrict__ esrc, const int* __restrict__ edst,
                                int n_edges, int n_nodes,
                                const float* __restrict__ h,
                                const float* __restrict__ as, const float* __restrict__ ad,
                                const unsigned* __restrict__ menc,
                                const float* __restrict__ den,
                                float* __restrict__ out) {
    int wid = (int)((blockIdx.x * blockDim.x + threadIdx.x) >> 5);
    int total = n_edges + n_nodes;
    if (wid >= total) return;
    int lane = threadIdx.x & 31;
    int s, d;
    if (wid < n_edges) { s = esrc[wid]; d = edst[wid]; } else { s = d = wid - n_edges; }
    float e = leaky(as[s] + ad[d]);
    float alpha = __expf(e - fdec(menc[d])) / (den[d] + GNN_EPS);
    float v = h[(size_t)s * GNN_HID + lane];
    atomicAdd(&out[(size_t)d * GNN_HID + lane], v * alpha);
}

__global__ void gnn_bias_relu(float* __restrict__ x, const float* __restrict__ b,
                              int n, int f) {
    int t = blockIdx.x * blockDim.x + threadIdx.x;
    if (t >= n * f) return;
    float v = x[t] + b[t % f];
    x[t] = v > 0.0f ? v : 0.0f;
}

__global__ void gnn_final_linear(const float* __restrict__ h,
                                 const float* __restrict__ w,
                                 const float* __restrict__ b,
                                 float* __restrict__ out, int n) {
    int t = blockIdx.x * blockDim.x + threadIdx.x;
    if (t >= n) return;
    const float* r = h + (size_t)t * GNN_HID;
    float acc = 0.0f;
#pragma unroll
    for (int c = 0; c < GNN_HID; ++c) acc += r[c] * w[c];
    out[t] = acc + b[0];
}

// ---------------------------------------------------------------------------

static inline long long cdivll(long long a, long long b) { return (a + b - 1) / b; }

extern "C" void kernel_launch(void* const* d_in, const int* in_sizes, int n_in,
                              void* d_out, int out_size, void* d_ws, size_t ws_size,
                              hipStream_t stream) {
    (void)in_sizes; (void)n_in; (void)out_size; (void)ws_size;
    const float* x      = (const float*)d_in[0];
    const int*   eidx   = (const int*)d_in[1];        // [2, E] int32
    const float* W1     = (const float*)d_in[3];      // [64, 128]
    const float* a_src1 = (const float*)d_in[4];      // [4, 32]
    const float* a_dst1 = (const float*)d_in[5];
    const float* b1     = (const float*)d_in[6];      // [128]
    const float* W2     = (const float*)d_in[7];      // [128, 32]
    const float* a_src2 = (const float*)d_in[8];      // [1, 32]
    const float* a_dst2 = (const float*)d_in[9];
    const float* b2     = (const float*)d_in[10];     // [32]
    const float* W_lin  = (const float*)d_in[11];     // [32, 1]
    const float* b_lin  = (const float*)d_in[12];     // [1]
    float* out = (float*)d_out;

    const int N = GNN_N_NODES, E = GNN_N_EDGES;
    const int* esrc = eidx;
    const int* edst = eidx + E;

    // workspace partition (256B aligned slabs)
    char*  base = (char*)d_ws;
    size_t off  = 0;
    auto alloc = [&](size_t bytes) -> char* {
        char* p = base + off;
        off += (bytes + 255) & ~(size_t)255;
        return p;
    };
    float*    h1    = (float*)   alloc((size_t)N * 128 * 4);
    float*    as1   = (float*)   alloc((size_t)N * 4 * 4);
    float*    ad1   = (float*)   alloc((size_t)N * 4 * 4);
    unsigned* menc1 = (unsigned*)alloc((size_t)N * 4 * 4);
    float*    den1  = (float*)   alloc((size_t)N * 4 * 4);
    float*    agg1  = (float*)   alloc((size_t)N * 128 * 4);
    float*    h2    = (float*)   alloc((size_t)N * 32 * 4);
    float*    as2   = (float*)   alloc((size_t)N * 4);
    float*    ad2   = (float*)   alloc((size_t)N * 4);
    unsigned* menc2 = (unsigned*)alloc((size_t)N * 4);
    float*    den2  = (float*)   alloc((size_t)N * 4);
    float*    agg2  = (float*)   alloc((size_t)N * 32 * 4);

    const int BLK = 256;
    long long g;

    // ---- Layer 1 ----
    // h1 = x @ W1   (M=100000, K=64, N=128) -> 50000 waves
    g = cdivll((long long)(N / 16) * (128 / 16) * 32, BLK);
    gnn_gemm_wmma_f32<<<(int)g, BLK, 0, stream>>>(x, W1, h1, N, GNN_F_IN, GNN_HEADS * GNN_HID);

    g = cdivll((long long)N * GNN_HEADS, BLK);
    gnn_attn_coef<<<(int)g, BLK, 0, stream>>>(h1, a_src1, a_dst1, as1, ad1, N, GNN_HEADS, GNN_HID);

    g = cdivll((long long)N * 128, BLK);
    gnn_init_state<<<(int)g, BLK, 0, stream>>>(menc1, den1, agg1, N * GNN_HEADS, N * 128);

    g = cdivll((long long)(E + N) * GNN_HEADS, BLK);
    gnn_edge_max  <<<(int)g, BLK, 0, stream>>>(esrc, edst, E, N, GNN_HEADS, as1, ad1, menc1);
    gnn_edge_denom<<<(int)g, BLK, 0, stream>>>(esrc, edst, E, N, GNN_HEADS, as1, ad1, menc1, den1);

    g = cdivll((long long)(E + N) * 32, BLK);
    gnn_edge_agg_h4<<<(int)g, BLK, 0, stream>>>(esrc, edst, E, N, h1, as1, ad1, menc1, den1, agg1);

    g = cdivll((long long)N * 128, BLK);
    gnn_bias_relu<<<(int)g, BLK, 0, stream>>>(agg1, b1, N, 128);

    // ---- Layer 2 ----
    // h2 = agg1 @ W2  (M=100000, K=128, N=32) -> 12500 waves
    g = cdivll((long long)(N / 16) * (32 / 16) * 32, BLK);
    gnn_gemm_wmma_f32<<<(int)g, BLK, 0, stream>>>(agg1, W2, h2, N, GNN_HEADS * GNN_HID, GNN_HID);

    g = cdivll((long long)N, BLK);
    gnn_attn_coef<<<(int)g, BLK, 0, stream>>>(h2, a_src2, a_dst2, as2, ad2, N, 1, GNN_HID);

    g = cdivll((long long)N * 32, BLK);
    gnn_init_state<<<(int)g, BLK, 0, stream>>>(menc2, den2, agg2, N, N * 32);

    g = cdivll((long long)(E + N), BLK);
    gnn_edge_max  <<<(int)g, BLK, 0, stream>>>(esrc, edst, E, N, 1, as2, ad2, menc2);
    gnn_edge_denom<<<(int)g, BLK, 0, stream>>>(esrc, edst, E, N, 1, as2, ad2, menc2, den2);

    g = cdivll((long long)(E + N) * 32, BLK);
    gnn_edge_agg_h1<<<(int)g, BLK, 0, stream>>>(esrc, edst, E, N, h2, as2, ad2, menc2, den2, agg2);

    g = cdivll((long long)N * 32, BLK);
    gnn_bias_relu<<<(int)g, BLK, 0, stream>>>(agg2, b2, N, 32);

    // ---- Head ----
    g = cdivll((long long)N, BLK);
    gnn_final_linear<<<(int)g, BLK, 0, stream>>>(agg2, W_lin, b_lin, out, N);
}